// XdGate_549755814197
// MI455X (gfx1250) — compile-verified
//
#include <hip/hip_runtime.h>
#include <hip/hip_bf16.h>

typedef __attribute__((ext_vector_type(2))) float v2f;
typedef __attribute__((ext_vector_type(8))) float v8f;

// out = (I_27 (x) M (x) I_81) · x, with x: 3^8 = 6561 f32 and M a 3x3 0/1
// permutation matrix. Batched (3x3)·(3x81) GEMM mapped onto
// V_WMMA_F32_16X16X4_F32 tiles:
//   A = M zero-padded to 16x4 (K=3 pad column zero)
//   B = 4x16 column tile of x[o] (K-rows 0..2 valid, K=3 pad row zero)
//   D rows 0..2 are the exact result (0/1 weights -> bit-exact f32).
__global__ __launch_bounds__(32)
void xd_gate_wmma(const float* __restrict__ x,
                  const float* __restrict__ Mbuf,
                  float* __restrict__ out) {
  const int  lane = threadIdx.x;   // 0..31 (wave32)
  const int  mrow = lane & 15;
  const bool hi   = lane >= 16;

  // ---- A matrix 16x4: rows 0..2 = M, everything else zero ----
  // ISA layout: VGPR0 (a.x) holds K = hi?2:0 ; VGPR1 (a.y) holds K = hi?3:1
  float ax = 0.f, ay = 0.f;
  if (mrow < 3) {
    if (!hi) { ax = Mbuf[mrow * 3 + 0]; ay = Mbuf[mrow * 3 + 1]; }
    else     { ax = Mbuf[mrow * 3 + 2]; /* K=3 pad stays 0 */ }
  }
  v2f a; a.x = ax; a.y = ay;

  // ---- B matrix 4x16: column tile of x reshaped [27, 3, 81] ----
  const int o   = blockIdx.x / 6;          // outer block (sites 0..2), 0..26
  const int t   = blockIdx.x % 6;          // column tile over inner dim 81
  const int n0  = (t < 5) ? t * 16 : 65;   // last tile overlaps: same values rewritten
  const int col = n0 + mrow;               // 0..80, always in bounds
  const float* xb = x + o * 243;

  float bx  = xb[(hi ? 162 : 0) + col];    // K = hi?2:0 row
  float by1 = xb[81 + col];                // K = 1 row
  v2f b; b.x = bx; b.y = hi ? 0.f : by1;   // K = 3 pad row -> 0

  v8f c = {};
  v8f d = __builtin_amdgcn_wmma_f32_16x16x4_f32(
      /*neg_a=*/false, a, /*neg_b=*/false, b,
      /*c_mod=*/(short)0, c, /*reuse_a=*/false, /*reuse_b=*/false);

  // ---- store D rows 0..2 (row m in d[m]; lanes 0..15 carry N = lane) ----
  if (!hi) {
    float* ob = out + o * 243 + col;
    ob[0]   = d[0];
    ob[81]  = d[1];
    ob[162] = d[2];
  }
}

extern "C" void kernel_launch(void* const* d_in, const int* in_sizes, int n_in,
                              void* d_out, int out_size, void* d_ws, size_t ws_size,
                              hipStream_t stream) {
  (void)in_sizes; (void)n_in; (void)out_size; (void)d_ws; (void)ws_size;
  const float* x = (const float*)d_in[0];   // 6561 f32 state vector
  const float* M = (const float*)d_in[1];   // 9 f32 (3x3 permutation matrix)
  float* out = (float*)d_out;               // 6561 f32
  // 27 outer blocks x 6 column tiles = 162 single-wave workgroups,
  // one V_WMMA_F32_16X16X4_F32 each. Problem is ~52 KB total traffic:
  // launch-latency bound on MI455X, so one pass with zero redundant bytes
  // plus the matrix pipe is the right shape.
  xd_gate_wmma<<<27 * 6, 32, 0, stream>>>(x, M, out);
}